// CrossAttentionFiLM_53704271069780
// MI455X (gfx1250) — compile-verified
//
#include <hip/hip_runtime.h>
#include <math.h>

#define B_    1024
#define T_    128
#define F_    1024
#define Z_    256
#define TX_   768
#define H_    8
#define D_    128
#define EPSV  1e-5f
#define SCALE 0.08838834764831845f  /* 1/sqrt(128) */
#define LDP   1024

typedef float v2f __attribute__((ext_vector_type(2)));
typedef float v8f __attribute__((ext_vector_type(8)));
typedef unsigned int u32x4 __attribute__((ext_vector_type(4)));
typedef int i32x4 __attribute__((ext_vector_type(4)));
typedef int i32x8 __attribute__((ext_vector_type(8)));

__device__ __forceinline__ v8f wmma4(v2f a, v2f b, v8f c) {
  // D = A(16x4 f32) x B(4x16 f32) + C(16x16 f32)
  return __builtin_amdgcn_wmma_f32_16x16x4_f32(false, a, false, b, (short)0, c,
                                               false, false);
}

// Software-pipelined 4-tile GEMM: C[t] += X(16xK) x W_t^T, B rows contiguous.
__device__ __forceinline__ void gemm4_xwT(const float* xrow, const float* w0,
                                          const float* w1, const float* w2,
                                          const float* w3, int K, v8f c[4]) {
  v2f a  = *(const v2f*)(xrow);
  v2f b0 = *(const v2f*)(w0), b1 = *(const v2f*)(w1);
  v2f b2 = *(const v2f*)(w2), b3 = *(const v2f*)(w3);
  for (int k = 0; k < K - 4; k += 4) {
    v2f an = *(const v2f*)(xrow + k + 4);
    v2f n0 = *(const v2f*)(w0 + k + 4);
    v2f n1 = *(const v2f*)(w1 + k + 4);
    v2f n2 = *(const v2f*)(w2 + k + 4);
    v2f n3 = *(const v2f*)(w3 + k + 4);
    c[0] = wmma4(a, b0, c[0]);
    c[1] = wmma4(a, b1, c[1]);
    c[2] = wmma4(a, b2, c[2]);
    c[3] = wmma4(a, b3, c[3]);
    a = an; b0 = n0; b1 = n1; b2 = n2; b3 = n3;
  }
  c[0] = wmma4(a, b0, c[0]);
  c[1] = wmma4(a, b1, c[1]);
  c[2] = wmma4(a, b2, c[2]);
  c[3] = wmma4(a, b3, c[3]);
}

// Software-pipelined single-tile GEMM with two interleaved accumulators.
// K must be a multiple of 8.
__device__ __forceinline__ v8f gemm1_xwT(const float* xrow, const float* wrow,
                                         int K) {
  v8f c0 = {}, c1 = {};
  v2f a0 = *(const v2f*)(xrow),     b0 = *(const v2f*)(wrow);
  v2f a1 = *(const v2f*)(xrow + 4), b1 = *(const v2f*)(wrow + 4);
  for (int k = 0; k < K - 8; k += 8) {
    v2f an0 = *(const v2f*)(xrow + k + 8),  bn0 = *(const v2f*)(wrow + k + 8);
    v2f an1 = *(const v2f*)(xrow + k + 12), bn1 = *(const v2f*)(wrow + k + 12);
    c0 = wmma4(a0, b0, c0);
    c1 = wmma4(a1, b1, c1);
    a0 = an0; b0 = bn0; a1 = an1; b1 = bn1;
  }
  c0 = wmma4(a0, b0, c0);
  c1 = wmma4(a1, b1, c1);
  return c0 + c1;
}

// ---------------------------------------------------------------------------
// K0: gamma = z @ Wg^T + bg ; beta = z @ Wb^T + bb        [B,F]
// ---------------------------------------------------------------------------
__global__ __launch_bounds__(128) void k_film(const float* __restrict__ z,
    const float* __restrict__ Wg, const float* __restrict__ bg,
    const float* __restrict__ Wb, const float* __restrict__ bb,
    float* __restrict__ gamma, float* __restrict__ beta) {
  const int lane = threadIdx.x & 31, wave = threadIdx.x >> 5;
  const int l16 = lane & 15, lh = lane >> 4;
  const int m0 = blockIdx.x * 16;
  const int n0 = blockIdx.y * 256 + wave * 64;
  const float* xrow = z + (m0 + l16) * Z_ + 2 * lh;
  v8f cg[4] = {}, cb[4] = {};
  gemm4_xwT(xrow, Wg + (n0 + 0 + l16) * Z_ + 2 * lh,
                  Wg + (n0 + 16 + l16) * Z_ + 2 * lh,
                  Wg + (n0 + 32 + l16) * Z_ + 2 * lh,
                  Wg + (n0 + 48 + l16) * Z_ + 2 * lh, Z_, cg);
  gemm4_xwT(xrow, Wb + (n0 + 0 + l16) * Z_ + 2 * lh,
                  Wb + (n0 + 16 + l16) * Z_ + 2 * lh,
                  Wb + (n0 + 32 + l16) * Z_ + 2 * lh,
                  Wb + (n0 + 48 + l16) * Z_ + 2 * lh, Z_, cb);
#pragma unroll
  for (int t = 0; t < 4; ++t) {
    int col = n0 + t * 16 + l16;
#pragma unroll
    for (int r = 0; r < 8; ++r) {
      int row = m0 + r + 8 * lh;
      gamma[row * F_ + col] = cg[t][r] + bg[col];
      beta [row * F_ + col] = cb[t][r] + bb[col];
    }
  }
}

// ---------------------------------------------------------------------------
// K1: out = FiLM(LN(x @ Wl^T + bl)) ; q = out @ Wq^T + bq   (fused via LDS)
// ---------------------------------------------------------------------------
__global__ __launch_bounds__(512) void k_out_q(const float* __restrict__ x,
    const float* __restrict__ Wl, const float* __restrict__ bl,
    const float* __restrict__ ln1g, const float* __restrict__ ln1b,
    const float* __restrict__ gamma, const float* __restrict__ beta,
    const float* __restrict__ Wq, const float* __restrict__ bq,
    float* __restrict__ outb, float* __restrict__ qb) {
  __shared__ float s[16 * LDP];  // 64 KB tile of 16 rows x 1024
  const int lane = threadIdx.x & 31, wave = threadIdx.x >> 5;
  const int l16 = lane & 15, lh = lane >> 4;
  const int m0 = blockIdx.x * 16;
  const int n0 = wave * 64;  // 16 waves cover 1024 columns

  {  // lin = x @ Wl^T + bl  -> LDS
    const float* xrow = x + (m0 + l16) * F_ + 2 * lh;
    v8f c[4] = {};
    gemm4_xwT(xrow, Wl + (n0 + 0 + l16) * F_ + 2 * lh,
                    Wl + (n0 + 16 + l16) * F_ + 2 * lh,
                    Wl + (n0 + 32 + l16) * F_ + 2 * lh,
                    Wl + (n0 + 48 + l16) * F_ + 2 * lh, F_, c);
#pragma unroll
    for (int t = 0; t < 4; ++t) {
      int col = n0 + t * 16 + l16;
#pragma unroll
      for (int r = 0; r < 8; ++r)
        s[(r + 8 * lh) * LDP + col] = c[t][r] + bl[col];
    }
  }
  __syncthreads();
  {  // LayerNorm + FiLM: wave w owns row w (wave32 shfl reductions)
    float* srow = s + wave * LDP;
    float p = 0.f;
    for (int i = lane; i < F_; i += 32) p += srow[i];
#pragma unroll
    for (int o = 16; o; o >>= 1) p += __shfl_xor(p, o);
    float mu = p * (1.0f / F_);
    float pv = 0.f;
    for (int i = lane; i < F_; i += 32) { float d = srow[i] - mu; pv += d * d; }
#pragma unroll
    for (int o = 16; o; o >>= 1) pv += __shfl_xor(pv, o);
    float inv = rsqrtf(pv * (1.0f / F_) + EPSV);
    const int gr = m0 + wave;
    for (int i = lane; i < F_; i += 32) {
      float v = (srow[i] - mu) * inv * ln1g[i] + ln1b[i];
      v = v * (1.0f + gamma[gr * F_ + i]) + beta[gr * F_ + i];
      srow[i] = v;
      outb[gr * F_ + i] = v;
    }
  }
  __syncthreads();
  {  // q = out @ Wq^T + bq  (A-fragments straight from LDS)
    const float* arow = s + l16 * LDP + 2 * lh;
    v8f c[4] = {};
    gemm4_xwT(arow, Wq + (n0 + 0 + l16) * F_ + 2 * lh,
                    Wq + (n0 + 16 + l16) * F_ + 2 * lh,
                    Wq + (n0 + 32 + l16) * F_ + 2 * lh,
                    Wq + (n0 + 48 + l16) * F_ + 2 * lh, F_, c);
#pragma unroll
    for (int t = 0; t < 4; ++t) {
      int col = n0 + t * 16 + l16;
#pragma unroll
      for (int r = 0; r < 8; ++r)
        qb[(m0 + r + 8 * lh) * F_ + col] = c[t][r] + bq[col];
    }
  }
}

// ---------------------------------------------------------------------------
// K2: qk[b,h,c] = sum_dd q[b,h*128+dd] * Wk[h*128+dd, c]   (fold q into Wk)
// ---------------------------------------------------------------------------
__global__ __launch_bounds__(128) void k_qk(const float* __restrict__ qb,
    const float* __restrict__ Wk, float* __restrict__ qk) {
  const int lane = threadIdx.x & 31, wave = threadIdx.x >> 5;
  const int l16 = lane & 15, lh = lane >> 4;
  const int m0 = blockIdx.x * 16;
  const int h  = blockIdx.y;
  const int n0 = blockIdx.z * 256 + wave * 64;
  const float* xrow = qb + (m0 + l16) * F_ + h * D_ + 2 * lh;
  const float* wc   = Wk + (h * D_ + 2 * lh) * TX_ + n0 + l16;  // K-major B
  v8f c[4] = {};
  v2f a = *(const v2f*)(xrow);
  v2f bf[4];
#pragma unroll
  for (int t = 0; t < 4; ++t) { bf[t][0] = wc[t * 16]; bf[t][1] = wc[t * 16 + TX_]; }
  for (int k = 0; k < D_ - 4; k += 4) {
    const float* wp = wc + (k + 4) * TX_;
    v2f an = *(const v2f*)(xrow + k + 4);
    v2f bn[4];
#pragma unroll
    for (int t = 0; t < 4; ++t) { bn[t][0] = wp[t * 16]; bn[t][1] = wp[t * 16 + TX_]; }
#pragma unroll
    for (int t = 0; t < 4; ++t) c[t] = wmma4(a, bf[t], c[t]);
    a = an;
#pragma unroll
    for (int t = 0; t < 4; ++t) bf[t] = bn[t];
  }
#pragma unroll
  for (int t = 0; t < 4; ++t) c[t] = wmma4(a, bf[t], c[t]);
#pragma unroll
  for (int t = 0; t < 4; ++t)
#pragma unroll
    for (int r = 0; r < 8; ++r)
      qk[((m0 + r + 8 * lh) * H_ + h) * TX_ + n0 + t * 16 + l16] = c[t][r];
}

// ---------------------------------------------------------------------------
// K3: scores[b,h,t] = (text[b,t,:] . qk[b,h,:]) / sqrt(d)   (HBM pass 1)
//     qk[b] (24 KB) staged into LDS via the Tensor Data Mover.
// ---------------------------------------------------------------------------
__global__ __launch_bounds__(256) void k_scores(const float* __restrict__ text,
    const float* __restrict__ qk, float* __restrict__ scores) {
  __shared__ float sqk[16 * TX_];  // 8 head rows + 8 zero rows (B-frag padding)
  const int b = blockIdx.x;

  if (threadIdx.x < 32) {  // wave 0 issues one TDM descriptor for qk[b]
    unsigned long long ga = (unsigned long long)(uintptr_t)(qk + (size_t)b * H_ * TX_);
    unsigned int lds = (unsigned int)(uintptr_t)(&sqk[0]);
    u32x4 g0;
    g0[0] = 1u;                                   // count=1, user descriptor
    g0[1] = lds;                                  // lds_addr (bytes)
    g0[2] = (unsigned int)(ga & 0xffffffffu);     // global_addr[31:0]
    g0[3] = (unsigned int)((ga >> 32) & 0x1ffffffu) | (2u << 30);  // type=2
    i32x8 g1;
    g1[0] = 0x00020000;            // workgroup_mask=0 (no cluster), data_size=4B
    g1[1] = (int)(6144u << 16);    // tensor_dim0 = 6144 elements (low 16)
    g1[2] = (int)(1u << 16);       // tensor_dim0 hi=0 | tensor_dim1 = 1
    g1[3] = (int)(6144u << 16);    // tensor_dim1 hi=0 | tile_dim0 = 6144
    g1[4] = 1;                     // tile_dim1 = 1, tile_dim2 = 0
    g1[5] = 6144;                  // tensor_dim0_stride (low 32)
    g1[6] = 0;                     // stride hi | tensor_dim1_stride lo
    g1[7] = 0;
    i32x4 zz = {0, 0, 0, 0};
#if __clang_major__ >= 23
    i32x8 z8 = {0, 0, 0, 0, 0, 0, 0, 0};
    __builtin_amdgcn_tensor_load_to_lds(g0, g1, zz, zz, z8, 0);
#else
    __builtin_amdgcn_tensor_load_to_lds(g0, g1, zz, zz, 0);
#endif
    __builtin_amdgcn_s_wait_tensorcnt(0);
  }
  for (int i = threadIdx.x; i < H_ * TX_; i += 256) sqk[H_ * TX_ + i] = 0.f;
  __syncthreads();

  const int lane = threadIdx.x & 31, wave = threadIdx.x >> 5;
  const int l16 = lane & 15, lh = lane >> 4;
  const int t0 = wave * 16;  // 8 waves cover T=128
  const float* xrow = text + (b * T_ + t0 + l16) * TX_ + 2 * lh;
  const float* brow = sqk + l16 * TX_ + 2 * lh;
  v8f c = gemm1_xwT(xrow, brow, TX_);
  if (l16 < 8) {
#pragma unroll
    for (int r = 0; r < 8; ++r)
      scores[(b * H_ + l16) * T_ + t0 + r + 8 * lh] = c[r] * SCALE;
  }
}

// ---------------------------------------------------------------------------
// K4: masked softmax over T per (b,h), in place. (bk shift cancels here.)
// ---------------------------------------------------------------------------
__global__ __launch_bounds__(256) void k_softmax(float* __restrict__ scores,
    const int* __restrict__ attention) {
  const int b = blockIdx.x;
  const int lane = threadIdx.x & 31, h = threadIdx.x >> 5;
  float* srow = scores + (b * H_ + h) * T_;
  const int* arow = attention + b * T_;
  float v[4]; float mx = -INFINITY;
#pragma unroll
  for (int i = 0; i < 4; ++i) {
    int t = lane + i * 32;
    float s = srow[t];
    if (arow[t] == 0) s = -INFINITY;
    v[i] = s; mx = fmaxf(mx, s);
  }
#pragma unroll
  for (int o = 16; o; o >>= 1) mx = fmaxf(mx, __shfl_xor(mx, o));
  float sum = 0.f;
#pragma unroll
  for (int i = 0; i < 4; ++i) { v[i] = expf(v[i] - mx); sum += v[i]; }
#pragma unroll
  for (int o = 16; o; o >>= 1) sum += __shfl_xor(sum, o);
  float inv = 1.0f / sum;
#pragma unroll
  for (int i = 0; i < 4; ++i) srow[lane + i * 32] = v[i] * inv;
}

// ---------------------------------------------------------------------------
// K5: tctx[b,h,c] = sum_t attn[b,h,t] * text[b,t,c]        (HBM pass 2)
// ---------------------------------------------------------------------------
__global__ __launch_bounds__(256) void k_tctx(const float* __restrict__ attn,
    const float* __restrict__ text, float* __restrict__ tctx) {
  const int b = blockIdx.x;
  const int lane = threadIdx.x & 31, wave = threadIdx.x >> 5;
  const int l16 = lane & 15, lh = lane >> 4;
  const float amask = (l16 < 8) ? 1.0f : 0.0f;  // pad M (heads) 8 -> 16
  const float* arow = attn + (b * H_ + (l16 & 7)) * T_ + 2 * lh;
  const int nb = wave * 96;  // 8 waves x 6 tiles x 16 = 768 cols
  const float* tc = text + (b * T_ + 2 * lh) * TX_ + nb + l16;
  v8f c[6] = {};
  v2f a = *(const v2f*)(arow); a *= amask;
  v2f bf[6];
#pragma unroll
  for (int t = 0; t < 6; ++t) { bf[t][0] = tc[t * 16]; bf[t][1] = tc[t * 16 + TX_]; }
  for (int k = 0; k < T_ - 4; k += 4) {
    const float* tp = tc + (k + 4) * TX_;
    v2f an = *(const v2f*)(arow + k + 4); an *= amask;
    v2f bn[6];
#pragma unroll
    for (int t = 0; t < 6; ++t) { bn[t][0] = tp[t * 16]; bn[t][1] = tp[t * 16 + TX_]; }
#pragma unroll
    for (int t = 0; t < 6; ++t) c[t] = wmma4(a, bf[t], c[t]);
    a = an;
#pragma unroll
    for (int t = 0; t < 6; ++t) bf[t] = bn[t];
  }
#pragma unroll
  for (int t = 0; t < 6; ++t) c[t] = wmma4(a, bf[t], c[t]);
  if (lh == 0) {  // rows 0..7 = heads
#pragma unroll
    for (int t = 0; t < 6; ++t)
#pragma unroll
      for (int r = 0; r < 8; ++r)
        tctx[(b * H_ + r) * TX_ + nb + t * 16 + l16] = c[t][r];
  }
}

// ---------------------------------------------------------------------------
// K6a: ctx[b, h*128+dd] = tctx[b,h,:] . Wv[h*128+dd, :] + bv  (softmax sums to 1)
// ---------------------------------------------------------------------------
__global__ __launch_bounds__(256) void k_ctx(const float* __restrict__ tctx,
    const float* __restrict__ Wv, const float* __restrict__ bv,
    float* __restrict__ ctx) {
  const int lane = threadIdx.x & 31, wave = threadIdx.x >> 5;
  const int l16 = lane & 15, lh = lane >> 4;
  const int m0 = blockIdx.x * 16;
  const int h  = blockIdx.y;
  const int f0 = h * D_ + wave * 16;  // 8 waves x 16 = 128 features of head h
  const float* xrow = tctx + ((m0 + l16) * H_ + h) * TX_ + 2 * lh;
  const float* wrow = Wv + (f0 + l16) * TX_ + 2 * lh;
  v8f c = gemm1_xwT(xrow, wrow, TX_);
#pragma unroll
  for (int r = 0; r < 8; ++r)
    ctx[(m0 + r + 8 * lh) * F_ + f0 + l16] = c[r] + bv[f0 + l16];
}

// ---------------------------------------------------------------------------
// K6b: attn_out = ctx @ Wo^T + bo
// ---------------------------------------------------------------------------
__global__ __launch_bounds__(128) void k_ogemm(const float* __restrict__ X,
    const float* __restrict__ W, const float* __restrict__ bias,
    float* __restrict__ out) {
  const int lane = threadIdx.x & 31, wave = threadIdx.x >> 5;
  const int l16 = lane & 15, lh = lane >> 4;
  const int m0 = blockIdx.x * 16;
  const int n0 = blockIdx.y * 256 + wave * 64;
  const float* xrow = X + (m0 + l16) * F_ + 2 * lh;
  v8f c[4] = {};
  gemm4_xwT(xrow, W + (n0 + 0 + l16) * F_ + 2 * lh,
                  W + (n0 + 16 + l16) * F_ + 2 * lh,
                  W + (n0 + 32 + l16) * F_ + 2 * lh,
                  W + (n0 + 48 + l16) * F_ + 2 * lh, F_, c);
#pragma unroll
  for (int t = 0; t < 4; ++t) {
    int col = n0 + t * 16 + l16;
#pragma unroll
    for (int r = 0; r < 8; ++r)
      out[(m0 + r + 8 * lh) * F_ + col] = c[t][r] + bias[col];
  }
}

// ---------------------------------------------------------------------------
// K7: y = gelu_exact(out + LN(attn_out))
// ---------------------------------------------------------------------------
__global__ __launch_bounds__(256) void k_final(const float* __restrict__ outb,
    const float* __restrict__ attn_out, const float* __restrict__ ln2g,
    const float* __restrict__ ln2b, float* __restrict__ y) {
  __shared__ float sred[8];
  const int b = blockIdx.x;
  const int lane = threadIdx.x & 31, wave = threadIdx.x >> 5;
  const float* arow = attn_out + b * F_;
  float loc[4]; float p = 0.f;
#pragma unroll
  for (int i = 0; i < 4; ++i) { loc[i] = arow[threadIdx.x + i * 256]; p += loc[i]; }
#pragma unroll
  for (int o = 16; o; o >>= 1) p += __shfl_xor(p, o);
  if (lane == 0) sred[wave] = p;
  __syncthreads();
  float tot = 0.f;
#pragma unroll
  for (int j = 0; j < 8; ++j) tot += sred[j];
  float mu = tot * (1.0f / F_);
  __syncthreads();
  float pv = 0.f;
#pragma unroll
  for (int i = 0; i < 4; ++i) { float d = loc[i] - mu; pv += d * d; }
#pragma unroll
  for (int o = 16; o; o >>= 1) pv += __shfl_xor(pv, o);
  if (lane == 0) sred[wave] = pv;
  __syncthreads();
  float vt = 0.f;
#pragma unroll
  for (int j = 0; j < 8; ++j) vt += sred[j];
  float inv = rsqrtf(vt * (1.0f / F_) + EPSV);
#pragma unroll
  for (int i = 0; i < 4; ++i) {
    int col = threadIdx.x + i * 256;
    float v = (loc[i] - mu) * inv * ln2g[col] + ln2b[col];
    float r = outb[b * F_ + col] + v;
    y[b * F_ + col] = 0.5f * r * (1.0f + erff(r * 0.7071067811865475f));
  }
}

// ---------------------------------------------------------------------------
extern "C" void kernel_launch(void* const* d_in, const int* in_sizes, int n_in,
                              void* d_out, int out_size, void* d_ws, size_t ws_size,
                              hipStream_t stream) {
  (void)in_sizes; (void)n_in; (void)out_size; (void)ws_size;
  const float* x    = (const float*)d_in[0];
  const float* z    = (const float*)d_in[1];
  const float* text = (const float*)d_in[2];
  const int*   attn = (const int*)  d_in[3];
  const float* Wg = (const float*)d_in[4],  *bg = (const float*)d_in[5];
  const float* Wb = (const float*)d_in[6],  *bb = (const float*)d_in[7];
  const float* Wl = (const float*)d_in[8],  *bl = (const float*)d_in[9];
  const float* g1 = (const float*)d_in[10], *b1 = (const float*)d_in[11];
  const float* Wq = (const float*)d_in[12], *bq = (const float*)d_in[13];
  const float* Wk = (const float*)d_in[14];  // bk (d_in[15]): cancels in softmax
  const float* Wv = (const float*)d_in[16], *bv = (const float*)d_in[17];
  const float* Wo = (const float*)d_in[18], *bo = (const float*)d_in[19];
  const float* g2 = (const float*)d_in[20], *b2 = (const float*)d_in[21];
  float* y = (float*)d_out;
  float* w = (float*)d_ws;

  const size_t BF = (size_t)B_ * F_;
  float* gamma  = w;
  float* beta   = w + BF;
  float* outb   = w + 2 * BF;
  float* qb     = w + 3 * BF;
  float* qk     = w + 4 * BF;                              // B*H*TX floats
  float* scores = w + 4 * BF + (size_t)B_ * H_ * TX_;      // B*H*T floats
  float* tctx   = qk;     // reuse: qk dead after k_scores
  float* ctx    = gamma;  // reuse: gamma dead after k_out_q
  float* aout   = beta;   // reuse: beta  dead after k_out_q

  k_film   <<<dim3(B_ / 16, F_ / 256),       128, 0, stream>>>(z, Wg, bg, Wb, bb, gamma, beta);
  k_out_q  <<<dim3(B_ / 16),                 512, 0, stream>>>(x, Wl, bl, g1, b1, gamma, beta, Wq, bq, outb, qb);
  k_qk     <<<dim3(B_ / 16, H_, TX_ / 256),  128, 0, stream>>>(qb, Wk, qk);
  k_scores <<<dim3(B_),                      256, 0, stream>>>(text, qk, scores);
  k_softmax<<<dim3(B_),                      256, 0, stream>>>(scores, attn);
  k_tctx   <<<dim3(B_),                      256, 0, stream>>>(scores, text, tctx);
  k_ctx    <<<dim3(B_ / 16, H_),             256, 0, stream>>>(tctx, Wv, bv, ctx);
  k_ogemm  <<<dim3(B_ / 16, F_ / 256),       128, 0, stream>>>(ctx, Wo, bo, aout);
  k_final  <<<dim3(B_),                      256, 0, stream>>>(outb, aout, g2, b2, y);
}